// CRF_73589969650117
// MI455X (gfx1250) — compile-verified
//
#include <hip/hip_runtime.h>
#include <hip/hip_bf16.h>

// CRF: forward (logsumexp scan -> f32 WMMA 16x16x4 via exp-trick, async-LDS
// double-buffered emissions), gold score, Viterbi (max-plus VALU, LDS
// backpointers + LDS backtrace, async-LDS emission streaming).
// B=256, S=2048, T=50 (48 tags + START/STOP). Pad T to 64 for WMMA.

#define CRF_B   256
#define CRF_S   2048
#define CRF_T   50
#define CRF_TP  64
#define CRF_START 48
#define CRF_STOP  49

typedef __attribute__((ext_vector_type(2))) float v2f;
typedef __attribute__((ext_vector_type(8))) float v8f;

__device__ __forceinline__ unsigned lds_addr_u32(const void* p) {
    // Generic LDS pointer: low 32 bits are the wave-relative LDS byte address.
    return (unsigned)(size_t)p;
}

__device__ __forceinline__ void async_copy_b64(unsigned lds, const float* g) {
    asm volatile("global_load_async_to_lds_b64 %0, %1, off"
                 :: "v"(lds), "v"(g) : "memory");
}
__device__ __forceinline__ void async_copy_b32(unsigned lds, const float* g) {
    asm volatile("global_load_async_to_lds_b32 %0, %1, off"
                 :: "v"(lds), "v"(g) : "memory");
}

// ---------------------------------------------------------------- prep: E = exp(trans) zero-padded to 64x64; zero accumulators
__global__ void crf_prep(const float* __restrict__ trans,
                         float* __restrict__ Ew,
                         float* __restrict__ accs) {
    int idx = blockIdx.x * blockDim.x + threadIdx.x;
    if (idx < 2) accs[idx] = 0.0f;
    for (int k = idx; k < CRF_TP * CRF_TP; k += gridDim.x * blockDim.x) {
        int i = k / CRF_TP, j = k % CRF_TP;
        float v = 0.0f;
        if (i < CRF_T && j < CRF_T) v = __expf(trans[i * CRF_T + j]);
        Ew[k] = v;   // exp(-inf)==0 for the pad region => exact padding
    }
}

// ---------------------------------------------------------------- forward: 16 batches per WG, 4 waves, each wave owns a 16-col tile
__global__ __launch_bounds__(128)
void crf_forward(const float* __restrict__ feats,
                 const unsigned char* __restrict__ mask,
                 const float* __restrict__ trans,
                 const float* __restrict__ Ew,
                 float* __restrict__ fwd_acc) {
    __shared__ __align__(16) float pL[16 * CRF_TP];      // state p[b][j]
    __shared__ __align__(16) float aL[16 * CRF_TP];      // A = exp(p - pmax)
    __shared__ __align__(16) float fbuf[2][16 * CRF_TP]; // emission double buffer
    __shared__ float pmaxL[16];
    __shared__ float mkL[16];

    const int tid  = threadIdx.x;
    const int wave = tid >> 5;
    const int lane = tid & 31;
    const int hi   = lane >> 4;          // lane half (ISA A/B/C layout)
    const int lcol = lane & 15;
    const int col  = wave * 16 + lcol;   // output column j in [0,64)
    const int b0   = blockIdx.x * 16;

    // Preload constant B-matrix (E) fragments: 16 K-chunks of 4, v2f each.
    // B layout (4x16): VGPR g holds rows (4kc + g + 2*hi) striped across lanes.
    v2f ebf[16];
#pragma unroll
    for (int kc = 0; kc < 16; ++kc) {
        int r0 = 4 * kc + 2 * hi;
        ebf[kc].x = Ew[r0 * CRF_TP + col];
        ebf[kc].y = Ew[(r0 + 1) * CRF_TP + col];
    }

    // p0 = feats[:,0,:] + trans[START,:]
    for (int idx = tid; idx < 16 * CRF_TP; idx += 128) {
        int m = idx >> 6, j = idx & 63;
        float v = -1e30f;
        if (j < CRF_T)
            v = feats[(size_t)(b0 + m) * CRF_S * CRF_T + j]
              + trans[CRF_START * CRF_T + j];
        pL[idx] = v;
    }

    // Prime async emission tile for t=1 into fbuf[1].
    // 16 rows x 50 floats = 400 B64 chunks; 4 per thread, uniform EXEC
    // (out-of-range chunk indices clamp to chunk 0 -> duplicate benign writes)
#pragma unroll
    for (int k = 0; k < 4; ++k) {
        int e  = tid + 128 * k;
        int ee = (e < 400) ? e : 0;
        int m = ee / 25, c = ee % 25;
        async_copy_b64(lds_addr_u32(&fbuf[1][m * CRF_TP + 2 * c]),
                       feats + ((size_t)(b0 + m) * CRF_S + 1) * CRF_T + 2 * c);
    }
    __syncthreads();

    const int r = tid >> 3;   // batch row 0..15 (8 threads per row)
    const int q = tid & 7;

    for (int t = 1; t < CRF_S; ++t) {
        // ---- issue async emission loads for next step (uniform 4/thread) ----
        {
            const int tn = (t + 1 < CRF_S) ? (t + 1) : (CRF_S - 1);
            float* dst = fbuf[(t + 1) & 1];
#pragma unroll
            for (int k = 0; k < 4; ++k) {
                int e  = tid + 128 * k;
                int ee = (e < 400) ? e : 0;
                int m = ee / 25, c = ee % 25;
                async_copy_b64(lds_addr_u32(&dst[m * CRF_TP + 2 * c]),
                               feats + ((size_t)(b0 + m) * CRF_S + tn) * CRF_T + 2 * c);
            }
        }

        // ---- per-row max (8 lanes/row, shfl tree within wave) ----
        float mx = -1e30f;
#pragma unroll
        for (int k = 0; k < 8; ++k) mx = fmaxf(mx, pL[r * CRF_TP + q * 8 + k]);
        mx = fmaxf(mx, __shfl_xor(mx, 1, 8));
        mx = fmaxf(mx, __shfl_xor(mx, 2, 8));
        mx = fmaxf(mx, __shfl_xor(mx, 4, 8));
        if (q == 0) {
            pmaxL[r] = mx;
            mkL[r]   = (float)mask[(size_t)(b0 + r) * CRF_S + t];
        }
        // ---- A = exp(p - pmax) (pad cols give exp(-1e30)=0: exact) ----
#pragma unroll
        for (int k = 0; k < 8; ++k) {
            int j = q * 8 + k;
            aL[r * CRF_TP + j] = __expf(pL[r * CRF_TP + j] - mx);
        }
        // current-step emission tile (issued last iteration) must be resident:
        // each wave issued exactly 4 async ops above; <=4 outstanding means
        // all prior-step transfers completed (async loads complete in order).
        asm volatile("s_wait_asynccnt 0x4" ::: "memory");
        __syncthreads();

        // ---- C(16x16) = A(16x64) x E(64x16) via chained WMMA f32 16x16x4 ----
        v8f c = {};
#pragma unroll
        for (int kc = 0; kc < 16; ++kc) {
            // A layout (16x4): lane row = lcol; elems K = 4kc+2hi, 4kc+2hi+1
            const v2f a = *(const v2f*)&aL[lcol * CRF_TP + 4 * kc + 2 * hi];
            c = __builtin_amdgcn_wmma_f32_16x16x4_f32(
                    false, a, false, ebf[kc], (short)0, c, false, false);
        }

        // ---- p_new[m][col] = f_t + pmax + log(dot); C layout: row m=g+8*hi
        const float* fcur = fbuf[t & 1];
#pragma unroll
        for (int g = 0; g < 8; ++g) {
            int m = g + 8 * hi;
            float nf = -1e30f;
            if (col < CRF_T) {
                float f  = fcur[m * CRF_TP + col];         // LDS, not HBM
                float lv = __logf(fmaxf(c[g], 1e-38f));
                nf = f + pmaxL[m] + lv;
            }
            float oldp = pL[m * CRF_TP + col];
            pL[m * CRF_TP + col] = (mkL[m] != 0.0f) ? nf : oldp;
        }
        __syncthreads();
    }

    // ---- fwd_b = logsumexp_i(p[b][i] + trans[i][STOP]); sum over batches ----
    {
        float mx = -1e30f;
#pragma unroll
        for (int k = 0; k < 8; ++k) {
            int i = q * 8 + k;
            float v = (i < CRF_T) ? pL[r * CRF_TP + i] + trans[i * CRF_T + CRF_STOP]
                                  : -1e30f;
            mx = fmaxf(mx, v);
        }
        mx = fmaxf(mx, __shfl_xor(mx, 1, 8));
        mx = fmaxf(mx, __shfl_xor(mx, 2, 8));
        mx = fmaxf(mx, __shfl_xor(mx, 4, 8));
        float sm = 0.0f;
#pragma unroll
        for (int k = 0; k < 8; ++k) {
            int i = q * 8 + k;
            if (i < CRF_T)
                sm += __expf(pL[r * CRF_TP + i] + trans[i * CRF_T + CRF_STOP] - mx);
        }
        sm += __shfl_xor(sm, 1, 8);
        sm += __shfl_xor(sm, 2, 8);
        sm += __shfl_xor(sm, 4, 8);
        if (q == 0) atomicAdd(fwd_acc, mx + __logf(sm));
    }
}

// ---------------------------------------------------------------- gold score
__global__ __launch_bounds__(256)
void crf_gold(const float* __restrict__ feats,
              const unsigned char* __restrict__ mask,
              const int* __restrict__ tags,
              const float* __restrict__ trans,
              float* __restrict__ gold_acc) {
    const int b = blockIdx.x, tid = threadIdx.x;
    float acc = 0.0f;
    int cnt = 0;
    for (int s = tid; s < CRF_S; s += 256) {
        unsigned char mk = mask[(size_t)b * CRF_S + s];
        int tg = tags[(size_t)b * CRF_S + s];
        int pv = (s == 0) ? CRF_START : tags[(size_t)b * CRF_S + s - 1];
        float emit = feats[(size_t)b * CRF_S * CRF_T + (size_t)s * CRF_T + tg];
        float tr = trans[pv * CRF_T + tg];
        if (mk) acc += emit + tr;
        cnt += mk;
    }
    __shared__ float accL[256];
    __shared__ int   cntL[256];
    accL[tid] = acc; cntL[tid] = cnt;
    __syncthreads();
    for (int off = 128; off > 0; off >>= 1) {
        if (tid < off) { accL[tid] += accL[tid + off]; cntL[tid] += cntL[tid + off]; }
        __syncthreads();
    }
    if (tid == 0) {
        int len = cntL[0];
        int endid = tags[(size_t)b * CRF_S + (len - 1)];
        atomicAdd(gold_acc, accL[0] + trans[endid * CRF_T + CRF_STOP]);
    }
}

// ---------------------------------------------------------------- Viterbi: 1 batch/WG, backpointers in 128KB dynamic LDS
__global__ __launch_bounds__(64)
void crf_viterbi(const float* __restrict__ feats,
                 const unsigned char* __restrict__ mask,
                 const float* __restrict__ trans,
                 float* __restrict__ out_decode) {
    extern __shared__ unsigned char bpL[];          // [S][64] backpointers
    __shared__ float pl[CRF_TP];
    __shared__ __align__(8) float fbufV[2][CRF_TP]; // emission double buffer
    __shared__ int cntL[64];
    __shared__ int lenS;

    const int j = threadIdx.x;                      // 0..63 (col)
    const int b = blockIdx.x;
    const int jj = (j < CRF_T) ? j : 0;             // uniform-EXEC clamp

    float tcol[CRF_T];                              // trans column j in VGPRs
#pragma unroll
    for (int i = 0; i < CRF_T; ++i)
        tcol[i] = (j < CRF_T) ? trans[i * CRF_T + j] : -1e30f;

    float v0 = -1e30f;
    if (j < CRF_T)
        v0 = feats[(size_t)b * CRF_S * CRF_T + j] + trans[CRF_START * CRF_T + j];
    pl[j] = v0;

    // prime emissions for t=1 (1 async B32 per thread, uniform)
    async_copy_b32(lds_addr_u32(&fbufV[1][jj]),
                   feats + ((size_t)b * CRF_S + 1) * CRF_T + jj);
    __syncthreads();

    for (int t = 1; t < CRF_S; ++t) {
        // issue next step's emissions (exactly 1 async op per wave per step)
        const int tn = (t + 1 < CRF_S) ? (t + 1) : (CRF_S - 1);
        async_copy_b32(lds_addr_u32(&fbufV[(t + 1) & 1][jj]),
                       feats + ((size_t)b * CRF_S + tn) * CRF_T + jj);

        float best = -1e30f; int arg = 0;
#pragma unroll
        for (int i = 0; i < CRF_T; ++i) {           // max-plus column update
            float v = pl[i] + tcol[i];
            if (v > best) { best = v; arg = i; }
        }
        unsigned char mk = mask[(size_t)b * CRF_S + t];
        float oldp = pl[j];
        asm volatile("s_wait_asynccnt 0x1" ::: "memory"); // t's tile resident
        __syncthreads();
        float nf = best + fbufV[t & 1][j];
        pl[j] = mk ? ((j < CRF_T) ? nf : -1e30f) : oldp;
        bpL[(size_t)t * 64 + j] = mk ? (unsigned char)arg : 0;
        __syncthreads();
    }

    // length = sum(mask)
    int cnt = 0;
    for (int s = j; s < CRF_S; s += 64) cnt += mask[(size_t)b * CRF_S + s];
    cntL[j] = cnt;
    __syncthreads();
    if (j == 0) {
        int tot = 0;
        for (int k = 0; k < 64; ++k) tot += cntL[k];
        lenS = tot;
    }
    __syncthreads();

    if (j == 0) {
        const int len = lenS;
        float best = -1e30f; int ptr = 0;
        for (int i = 0; i < CRF_T; ++i) {
            float v = pl[i] + trans[i * CRF_T + CRF_STOP];
            if (v > best) { best = v; ptr = i; }
        }
        out_decode[(size_t)b * CRF_S + (CRF_S - 1)] = (float)ptr;
        int cur = ptr;
        for (int s = CRF_S - 2; s >= 0; --s) {      // LDS-resident backtrace
            int val = (s == len - 1) ? ptr : (int)bpL[(size_t)(s + 1) * 64 + cur];
            out_decode[(size_t)b * CRF_S + s] = (float)val;
            cur = val;
        }
    }
}

// ---------------------------------------------------------------- finish
__global__ void crf_finish(const float* __restrict__ accs, float* __restrict__ out) {
    if (threadIdx.x == 0 && blockIdx.x == 0) out[0] = accs[0] - accs[1];
}

extern "C" void kernel_launch(void* const* d_in, const int* in_sizes, int n_in,
                              void* d_out, int out_size, void* d_ws, size_t ws_size,
                              hipStream_t stream) {
    const float*         feats = (const float*)d_in[0];
    const unsigned char* mask  = (const unsigned char*)d_in[1];
    const int*           tags  = (const int*)d_in[2];
    const float*         trans = (const float*)d_in[3];
    float* out  = (float*)d_out;
    float* ws   = (float*)d_ws;
    float* accs = ws;            // [0]=fwd score, [1]=gold score
    float* Ew   = ws + 64;       // 64x64 exp(trans), 16KB

    crf_prep<<<4, 256, 0, stream>>>(trans, Ew, accs);
    crf_forward<<<CRF_B / 16, 128, 0, stream>>>(feats, mask, trans, Ew, accs);
    crf_gold<<<CRF_B, 256, 0, stream>>>(feats, mask, tags, trans, accs + 1);
    crf_viterbi<<<CRF_B, 64, CRF_S * 64, stream>>>(feats, mask, trans, out + 1);
    crf_finish<<<1, 1, 0, stream>>>(accs, out);
}